// ScaledDotProductAttention_57801669869696
// MI455X (gfx1250) — compile-verified
//
#include <hip/hip_runtime.h>

typedef __attribute__((ext_vector_type(2))) float v2f;
typedef __attribute__((ext_vector_type(8))) float v8f;

constexpr int kB = 16;
constexpr int kS = 2048;
constexpr int kD = 128;
constexpr int kQT = 16;            // query rows per wave
constexpr int kST = 2050;          // padded LDS row stride (floats); even -> 8B-aligned v2f, 2050%64==2 -> conflict-free scans
constexpr float kInvScale = 0.08838834764831845f;   // 1/sqrt(128)
constexpr float kNegBig   = -3.402823466e38f;       // finfo(f32).min

extern "C" __global__ __launch_bounds__(32)
void sdpa_wmma_f32(const float* __restrict__ Q,
                   const float* __restrict__ K,
                   const float* __restrict__ V,
                   const unsigned char* __restrict__ mask,
                   float* __restrict__ Out,
                   float* __restrict__ Attn)
{
    extern __shared__ float smem[];
    float* sc   = smem;                 // 16 x 2050 score/P tile
    float* rmax = smem + kQT * kST;     // 16 row maxima
    float* rsum = rmax + kQT;           // 16 row sums

    const int lane = threadIdx.x;       // 0..31 (wave32)
    const int lo   = lane & 15;
    const int hi   = lane >> 4;

    const int blk = blockIdx.x;
    const int b   = blk >> 7;           // / (S/16)
    const int qt  = blk & 127;
    const int q0  = qt * 16;

    const float* Qb = Q + (size_t)b * kS * kD;
    const float* Kb = K + (size_t)b * kS * kD;
    const float* Vb = V + (size_t)b * kS * kD;
    const unsigned char* mb = mask + (size_t)b * kS;
    float* outb  = Out  + ((size_t)b * kS + q0) * kD;
    float* attnb = Attn + ((size_t)b * kS + q0) * kS;

    // ---- query_has_keys: OR-reduce mask over all keys of this batch ----
    int any = 0;
    for (int j = lane; j < kS; j += 32) any |= mb[j];
    for (int off = 16; off >= 1; off >>= 1) any |= __shfl_xor(any, off, 32);

    if (!any) {
        // every key masked: weights and outputs are exactly zero
        for (int idx = lane; idx < kQT * kD; idx += 32) outb[idx] = 0.0f;
        for (int m = 0; m < kQT; ++m)
            for (int j = lane; j < kS; j += 32)
                attnb[(size_t)m * kS + j] = 0.0f;
        return;
    }

    // ---- load Q tile (16x128) into V_WMMA_F32_16X16X4_F32 A-fragments ----
    // lanes 0-15: row M=lane, holds K=0,1 ; lanes 16-31: row M=lane-16, holds K=2,3
    // FULLY unrolled everywhere it is used so afrag[] stays in VGPRs (no scratch!)
    v2f afrag[32];
    {
        const float* qrow = Qb + (size_t)(q0 + lo) * kD + 2 * hi;   // 8B aligned
        #pragma unroll
        for (int c = 0; c < 32; ++c)
            afrag[c] = *(const v2f*)(qrow + 4 * c);
    }

    // ================= S = Q K^T / sqrt(d), masked, into LDS =================
    for (int kt = 0; kt < kS / 16; ++kt) {
        const float* krow = Kb + (size_t)(kt * 16 + lo) * kD + 2 * hi;
        // prefetch next key tile (speculative; dropped if OOB)
        __builtin_prefetch(krow + kD * 16, 0, 1);

        v8f acc = {0.f, 0.f, 0.f, 0.f, 0.f, 0.f, 0.f, 0.f};
        #pragma unroll
        for (int c = 0; c < 32; ++c) {
            v2f bfrag = *(const v2f*)(krow + 4 * c);   // B: 4x16, col N=lo, K rows 2*hi{+0,1}
            acc = __builtin_amdgcn_wmma_f32_16x16x4_f32(
                false, afrag[c], false, bfrag, (short)0, acc, false, false);
        }

        // C/D layout: VGPR r, lanes 0-15 -> (M=r, N=lane); lanes 16-31 -> (M=r+8, N=lane-16)
        const bool allow = mb[kt * 16 + lo] != 0;
        #pragma unroll
        for (int r = 0; r < 8; ++r) {
            const int m = r + 8 * hi;
            sc[m * kST + kt * 16 + lo] = allow ? acc[r] * kInvScale : kNegBig;
        }
    }
    __syncthreads();

    // ================= row softmax (cooperative, coalesced) =================
    for (int m = 0; m < kQT; ++m) {
        float mx = kNegBig;
        for (int j = lane; j < kS; j += 32) mx = fmaxf(mx, sc[m * kST + j]);
        for (int off = 16; off >= 1; off >>= 1) mx = fmaxf(mx, __shfl_xor(mx, off, 32));
        if (lane == 0) rmax[m] = mx;
    }
    __syncthreads();
    for (int m = 0; m < kQT; ++m) {
        const float mx = rmax[m];
        float sm = 0.0f;
        for (int j = lane; j < kS; j += 32) sm += __expf(sc[m * kST + j] - mx);
        for (int off = 16; off >= 1; off >>= 1) sm += __shfl_xor(sm, off, 32);
        if (lane == 0) rsum[m] = sm;
    }
    __syncthreads();
    for (int m = 0; m < kQT; ++m) {
        const float mx  = rmax[m];
        const float inv = 1.0f / rsum[m];
        for (int j = lane; j < kS; j += 32) {
            const float w = __expf(sc[m * kST + j] - mx) * inv;
            sc[m * kST + j] = w;                    // keep normalized P for P·V
            attnb[(size_t)m * kS + j] = w;          // coalesced 128B/wave stores
        }
    }
    __syncthreads();

    // ================= O = P V (16x2048 · 2048x128) =================
    v8f oacc[8];
    #pragma unroll
    for (int nt = 0; nt < 8; ++nt) oacc[nt] = (v8f){0.f,0.f,0.f,0.f,0.f,0.f,0.f,0.f};

    const float* prow = sc + lo * kST + 2 * hi;     // A fragment source in LDS (8B aligned: even index)
    #pragma unroll 2
    for (int c = 0; c < kS / 4; ++c) {
        const v2f ap = *(const v2f*)(prow + 4 * c);             // one ds_load_b64
        const float* vrow = Vb + (size_t)(4 * c + 2 * hi) * kD + lo;
        #pragma unroll
        for (int nt = 0; nt < 8; ++nt) {
            v2f bp;
            bp.x = vrow[nt * 16];
            bp.y = vrow[nt * 16 + kD];
            oacc[nt] = __builtin_amdgcn_wmma_f32_16x16x4_f32(
                false, ap, false, bp, (short)0, oacc[nt], false, false);
        }
    }

    // ---- store O tile: lane -> (M=r+8*hi, d=nt*16+lo) ----
    #pragma unroll
    for (int nt = 0; nt < 8; ++nt) {
        #pragma unroll
        for (int r = 0; r < 8; ++r) {
            const int m = r + 8 * hi;
            outb[(size_t)m * kD + nt * 16 + lo] = oacc[nt][r];
        }
    }
}

extern "C" void kernel_launch(void* const* d_in, const int* in_sizes, int n_in,
                              void* d_out, int out_size, void* d_ws, size_t ws_size,
                              hipStream_t stream) {
    const float* Q = (const float*)d_in[0];
    const float* K = (const float*)d_in[1];
    const float* V = (const float*)d_in[2];
    const unsigned char* mask = (const unsigned char*)d_in[3];

    float* Out  = (float*)d_out;                       // (B,S,D) first
    float* Attn = Out + (size_t)kB * kS * kD;          // then (B,S,S)

    const size_t shmem = (size_t)(kQT * kST + 2 * kQT) * sizeof(float);
    dim3 grid(kB * (kS / 16));                          // 2048 waves, one 16-query tile each
    sdpa_wmma_f32<<<grid, 32, shmem, stream>>>(Q, K, V, mask, Out, Attn);
}